// Mixer_10557029614277
// MI455X (gfx1250) — compile-verified
//
#include <hip/hip_runtime.h>
#include <math.h>

typedef float v2f __attribute__((ext_vector_type(2)));
typedef float v8f __attribute__((ext_vector_type(8)));

#define BT    (128 * 200)
#define NA    8
#define NG    2
#define GNUM  4
#define EMB   32
#define HYP   64
#define AH    64
#define SDIM  128
#define TILE  16
#define WPB   4
#define SC_LD 68    /* padded LDS row stride (floats) -> conflict-free */
#define GS_LD 132

__device__ inline v8f wmma4(v2f a, v2f b, v8f c) {
  return __builtin_amdgcn_wmma_f32_16x16x4_f32(
      /*neg_a=*/false, a, /*neg_b=*/false, b,
      /*c_mod=*/(short)0, c, /*reuse_a=*/false, /*reuse_b=*/false);
}

// NT 16x16 output tiles of X[16 x K] (row-major, ldx) @ W[K x NT*16] (row-major,
// ldw). A fragment loaded once per k-step and reused across all NT column tiles;
// NT independent accumulator chains interleave between dependent WMMAs.
template <int NT>
__device__ inline void gemm_tiles(const float* X, int ldx,
                                  const float* W, int ldw,
                                  int K, v8f c[NT],
                                  int row, int koff, int col) {
#pragma unroll 2
  for (int k = 0; k < K; k += 4) {
    v2f a;
    a.x = X[row * ldx + k + koff];
    a.y = X[row * ldx + k + koff + 1];
#pragma unroll
    for (int t = 0; t < NT; ++t) {
      v2f b;
      b.x = W[(k + koff) * ldw + t * 16 + col];
      b.y = W[(k + koff + 1) * ldw + t * 16 + col];
      c[t] = wmma4(a, b, c[t]);
    }
  }
}

__global__ __launch_bounds__(128) void mixer_kernel(
    const float* __restrict__ qvals, const float* __restrict__ states,
    const float* __restrict__ a_h,   const float* __restrict__ gs,
    const float* __restrict__ W1a,  const float* __restrict__ b1a,
    const float* __restrict__ W1b,  const float* __restrict__ b1b,
    const float* __restrict__ Wb1,  const float* __restrict__ bb1,
    const float* __restrict__ Ww2a, const float* __restrict__ bw2a,
    const float* __restrict__ Ww2b, const float* __restrict__ bw2b,
    const float* __restrict__ Wb2a, const float* __restrict__ bb2a,
    const float* __restrict__ Wb2b, const float* __restrict__ bb2b,
    const float* __restrict__ W3a,  const float* __restrict__ b3a,
    const float* __restrict__ W3b,  const float* __restrict__ b3b,
    const float* __restrict__ Wb3,  const float* __restrict__ bb3,
    const float* __restrict__ W4a,  const float* __restrict__ b4a,
    const float* __restrict__ W4b,  const float* __restrict__ b4b,
    const float* __restrict__ Wb4a, const float* __restrict__ bb4a,
    const float* __restrict__ Wb4b, const float* __restrict__ bb4b,
    float* __restrict__ out_tot, float* __restrict__ out_sd,
    float* __restrict__ ws_partial) {
  const int wave = threadIdx.x >> 5;
  const int lane = threadIdx.x & 31;
  const int tile = blockIdx.x * WPB + wave;
  const int s0   = tile * TILE;
  const int row  = lane & 15;
  const int hi   = lane >> 4;
  const int koff = hi * 2;
  const int col  = row;

  __shared__ float sc_[WPB][TILE][SC_LD];
  __shared__ float gsm_[WPB][TILE][GS_LD];
  __shared__ float qt_[WPB][TILE][NA];
  __shared__ float misc_[WPB][TILE][4];
  float (*sc)[SC_LD] = sc_[wave];
  float (*gsm)[GS_LD] = gsm_[wave];
  float (*qt)[NA] = qt_[wave];
  float (*misc)[4] = misc_[wave];

  // ---- stage group_state (max over GN agents per group) and q-tile in LDS ----
  for (int idx = lane; idx < TILE * 2 * HYP; idx += 32) {
    int r = idx >> 7, rem = idx & 127;
    int g = rem >> 6, h = rem & 63;
    const float* p = gs + ((size_t)(s0 + r) * NA + g * GNUM) * HYP + h;
    float m = p[0];
    m = fmaxf(m, p[HYP]);
    m = fmaxf(m, p[2 * HYP]);
    m = fmaxf(m, p[3 * HYP]);
    gsm[r][g * HYP + h] = m;
  }
  for (int idx = lane; idx < TILE * NA; idx += 32)
    qt[idx >> 3][idx & 7] = qvals[(size_t)(s0 + (idx >> 3)) * NA + (idx & 7)];

  // =================== per-group hypernets + mixing ===================
  for (int g = 0; g < NG; ++g) {
    // b1 = (sum_{n in g} a_h_n) @ Wb1[g] + GN*bb1  -> init mixing accumulator
    for (int idx = lane; idx < TILE * AH; idx += 32) {
      int r = idx >> 6, d = idx & 63;
      const float* p = a_h + ((size_t)(s0 + r) * NA + g * GNUM) * AH + d;
      sc[r][d] = p[0] + p[AH] + p[2 * AH] + p[3 * AH];
    }
    v8f macc[2] = {};
    gemm_tiles<2>(&sc[0][0], SC_LD, Wb1 + g * AH * EMB, EMB, AH, macc,
                  row, koff, col);
    for (int t = 0; t < 2; ++t) {
      float bias = (float)GNUM * bb1[g * EMB + t * 16 + col];
      for (int r = 0; r < 8; ++r) macc[t][r] += bias;
    }

    // per-agent hypernet: h1 = relu(a_h@W1a+b1a); w1 = |h1@W1b+b1b|
    for (int n = g * GNUM; n < (g + 1) * GNUM; ++n) {
      const float* Xa = a_h + ((size_t)s0 * NA + n) * AH;
      {
        v8f ct[4] = {};
        gemm_tiles<4>(Xa, NA * AH, W1a + n * AH * HYP, HYP, AH, ct,
                      row, koff, col);
        for (int t = 0; t < 4; ++t) {
          float bias = b1a[n * HYP + t * 16 + col];
          for (int r = 0; r < 8; ++r)
            sc[hi * 8 + r][t * 16 + col] = fmaxf(ct[t][r] + bias, 0.f);
        }
      }
      float lsum = 0.f;
      {
        v8f ct[2] = {};
        gemm_tiles<2>(&sc[0][0], SC_LD, W1b + n * HYP * EMB, EMB, HYP, ct,
                      row, koff, col);
        for (int t = 0; t < 2; ++t) {
          float bias = b1b[n * EMB + t * 16 + col];
          for (int r = 0; r < 8; ++r) {
            float v = fabsf(ct[t][r] + bias);
            macc[t][r] += qt[hi * 8 + r][n] * v;  // q-weighted mix accumulation
            lsum += v;
          }
        }
      }
      if (g == 0) {  // w1_avg partials for group 0 (deterministic 2-pass mean)
        for (int off = 16; off > 0; off >>= 1) lsum += __shfl_xor(lsum, off, 32);
        if (lane == 0) ws_partial[(size_t)tile * GNUM + n] = lsum;
      }
    }

    // hw = relu(GS_g @ Ww2a[g] + bw2a) -> sc ; w2 = |sc @ Ww2b[g] + bw2b|
    {
      v8f ct[4] = {};
      gemm_tiles<4>(&gsm[0][0] + g * HYP, GS_LD, Ww2a + g * HYP * HYP, HYP,
                    HYP, ct, row, koff, col);
      for (int t = 0; t < 4; ++t) {
        float bias = bw2a[g * HYP + t * 16 + col];
        for (int r = 0; r < 8; ++r)
          sc[hi * 8 + r][t * 16 + col] = fmaxf(ct[t][r] + bias, 0.f);
      }
    }
    v8f w2t[2] = {};
    gemm_tiles<2>(&sc[0][0], SC_LD, Ww2b + g * HYP * EMB, EMB, HYP, w2t,
                  row, koff, col);
    for (int t = 0; t < 2; ++t) {
      float bias = bw2b[g * EMB + t * 16 + col];
      for (int r = 0; r < 8; ++r) w2t[t][r] = fabsf(w2t[t][r] + bias);
    }
    // hb = relu(GS_g @ Wb2a[g] + bb2a) -> sc cols 0..31
    {
      v8f ct[2] = {};
      gemm_tiles<2>(&gsm[0][0] + g * HYP, GS_LD, Wb2a + g * HYP * EMB, EMB,
                    HYP, ct, row, koff, col);
      for (int t = 0; t < 2; ++t) {
        float bias = bb2a[g * EMB + t * 16 + col];
        for (int r = 0; r < 8; ++r)
          sc[hi * 8 + r][t * 16 + col] = fmaxf(ct[t][r] + bias, 0.f);
      }
    }
    // hidden = elu(macc); products into sc cols 32..47 for column reduction
    for (int r = 0; r < 8; ++r) {
      float h0 = macc[0][r]; h0 = (h0 > 0.f) ? h0 : expm1f(h0);
      float h1 = macc[1][r]; h1 = (h1 > 0.f) ? h1 : expm1f(h1);
      sc[hi * 8 + r][32 + col] = h0 * w2t[0][r] + h1 * w2t[1][r];
    }
    if (lane < 16) {  // b2 = hb@Wb2b + bb2b ; q_group = <hidden,w2> + b2
      float b2v = bb2b[g];
      for (int e = 0; e < EMB; ++e) b2v += sc[lane][e] * Wb2b[g * EMB + e];
      float qg = b2v;
      for (int e = 0; e < 16; ++e) qg += sc[lane][32 + e];
      misc[lane][g] = qg;
    }
  }

  // =================== top-level mixing over groups ===================
  v8f thacc[2];
  for (int t = 0; t < 2; ++t) {
    float bias = 2.f * bb3[t * 16 + col];  // b3 = sum_g(GS_g@Wb3 + bb3)
    for (int r = 0; r < 8; ++r) thacc[t][r] = bias;
  }
  for (int g = 0; g < NG; ++g) {
    gemm_tiles<2>(&gsm[0][0] + g * HYP, GS_LD, Wb3, EMB, HYP, thacc,
                  row, koff, col);
    {  // h3 = relu(GS_g@W3a + b3a) -> sc
      v8f ct[4] = {};
      gemm_tiles<4>(&gsm[0][0] + g * HYP, GS_LD, W3a, HYP, HYP, ct,
                    row, koff, col);
      for (int t = 0; t < 4; ++t) {
        float bias = b3a[t * 16 + col];
        for (int r = 0; r < 8; ++r)
          sc[hi * 8 + r][t * 16 + col] = fmaxf(ct[t][r] + bias, 0.f);
      }
    }
    {  // w3 = |h3@W3b + b3b| ; thacc += qg*w3
      v8f ct[2] = {};
      gemm_tiles<2>(&sc[0][0], SC_LD, W3b, EMB, HYP, ct, row, koff, col);
      for (int t = 0; t < 2; ++t) {
        float bias = b3b[t * 16 + col];
        for (int r = 0; r < 8; ++r)
          thacc[t][r] += misc[hi * 8 + r][g] * fabsf(ct[t][r] + bias);
      }
    }
  }
  for (int t = 0; t < 2; ++t)
    for (int r = 0; r < 8; ++r) {
      float x = thacc[t][r];
      thacc[t][r] = (x > 0.f) ? x : expm1f(x);
    }

  // h4 = relu(states@W4a + b4a) -> sc ; w4 = |sc@W4b + b4b|
  const float* Xs = states + (size_t)s0 * SDIM;
  {
    v8f ct[4] = {};
    gemm_tiles<4>(Xs, SDIM, W4a, HYP, SDIM, ct, row, koff, col);
    for (int t = 0; t < 4; ++t) {
      float bias = b4a[t * 16 + col];
      for (int r = 0; r < 8; ++r)
        sc[hi * 8 + r][t * 16 + col] = fmaxf(ct[t][r] + bias, 0.f);
    }
  }
  v8f w4t[2] = {};
  gemm_tiles<2>(&sc[0][0], SC_LD, W4b, EMB, HYP, w4t, row, koff, col);
  for (int t = 0; t < 2; ++t) {
    float bias = b4b[t * 16 + col];
    for (int r = 0; r < 8; ++r) w4t[t][r] = fabsf(w4t[t][r] + bias);
  }
  // b4 pre-activation: relu(states @ Wb4a + bb4a) -> sc cols 0..31
  {
    v8f ct[2] = {};
    gemm_tiles<2>(Xs, SDIM, Wb4a, EMB, SDIM, ct, row, koff, col);
    for (int t = 0; t < 2; ++t) {
      float bias = bb4a[t * 16 + col];
      for (int r = 0; r < 8; ++r)
        sc[hi * 8 + r][t * 16 + col] = fmaxf(ct[t][r] + bias, 0.f);
    }
  }
  // products tot_hidden * w4 -> sc cols 32..47
  for (int r = 0; r < 8; ++r)
    sc[hi * 8 + r][32 + col] = thacc[0][r] * w4t[0][r] + thacc[1][r] * w4t[1][r];

  if (lane < 16) {
    float b4v = bb4b[0];
    for (int e = 0; e < EMB; ++e) b4v += sc[lane][e] * Wb4b[e];
    float tq = b4v;
    for (int e = 0; e < 16; ++e) tq += sc[lane][32 + e];
    out_tot[s0 + lane] = tq;
  }

  // ---- sd_loss: signed cosine-similarity sum, all 32 lanes active ----
  // lane L and L+16 handle the same sample, opposite 32-dim halves of the
  // 64-dim reductions; combined with one shfl_xor(16) each.
  {
    const int sr = lane & 15;
    const float* gr = gs + (size_t)(s0 + sr) * NA * HYP + hi * 32;
    float inv[NA];
    for (int n = 0; n < NA; ++n) {
      float s = 0.f;
      for (int h = 0; h < 32; ++h) { float v = gr[n * HYP + h]; s += v * v; }
      s += __shfl_xor(s, 16, 32);
      inv[n] = rsqrtf(s);
    }
    float sd = 0.f;
    for (int n = 0; n < NA; ++n)
      for (int m = n + 1; m < NA; ++m) {
        float d = 0.f;
        for (int h = 0; h < 32; ++h) d += gr[n * HYP + h] * gr[m * HYP + h];
        float sgn = ((n >> 2) == (m >> 2)) ? -1.f : 1.f;
        sd += sgn * d * inv[n] * inv[m];
      }
    sd += __shfl_xor(sd, 16, 32);       // off-diagonal upper triangle (full dots)
    sd = 2.f * sd - (float)NA;          // symmetrize + diagonal (cos=1, sign=-1)
    if (lane < 16) out_sd[s0 + sr] = sd * (1.f / (float)(NA * NA));
  }
}

__global__ void w1avg_reduce(const float* __restrict__ ws,
                             float* __restrict__ out, int ntiles) {
  int n = threadIdx.x;
  if (n < GNUM) {
    float s = 0.f;
    for (int t = 0; t < ntiles; ++t) s += ws[(size_t)t * GNUM + n];
    out[n] = s / (float)((size_t)BT * EMB);
  }
}

extern "C" void kernel_launch(void* const* d_in, const int* in_sizes, int n_in,
                              void* d_out, int out_size, void* d_ws, size_t ws_size,
                              hipStream_t stream) {
  (void)in_sizes; (void)n_in; (void)out_size; (void)ws_size;
  const float* qvals  = (const float*)d_in[0];
  const float* states = (const float*)d_in[1];
  const float* a_h    = (const float*)d_in[2];
  const float* gs     = (const float*)d_in[3];
  const float* W1a  = (const float*)d_in[4];  const float* b1a  = (const float*)d_in[5];
  const float* W1b  = (const float*)d_in[6];  const float* b1b  = (const float*)d_in[7];
  const float* Wb1  = (const float*)d_in[8];  const float* bb1  = (const float*)d_in[9];
  const float* Ww2a = (const float*)d_in[10]; const float* bw2a = (const float*)d_in[11];
  const float* Ww2b = (const float*)d_in[12]; const float* bw2b = (const float*)d_in[13];
  const float* Wb2a = (const float*)d_in[14]; const float* bb2a = (const float*)d_in[15];
  const float* Wb2b = (const float*)d_in[16]; const float* bb2b = (const float*)d_in[17];
  const float* W3a  = (const float*)d_in[18]; const float* b3a  = (const float*)d_in[19];
  const float* W3b  = (const float*)d_in[20]; const float* b3b  = (const float*)d_in[21];
  const float* Wb3  = (const float*)d_in[22]; const float* bb3  = (const float*)d_in[23];
  const float* W4a  = (const float*)d_in[24]; const float* b4a  = (const float*)d_in[25];
  const float* W4b  = (const float*)d_in[26]; const float* b4b  = (const float*)d_in[27];
  const float* Wb4a = (const float*)d_in[28]; const float* bb4a = (const float*)d_in[29];
  const float* Wb4b = (const float*)d_in[30]; const float* bb4b = (const float*)d_in[31];

  float* out      = (float*)d_out;
  float* out_tot  = out;                 // [bt]
  float* out_w1   = out + BT;            // [4]
  float* out_sd   = out + BT + GNUM;     // [bt]
  float* ws       = (float*)d_ws;        // [1600][4] partials

  const int ntiles = BT / TILE;          // 1600
  const int blocks = ntiles / WPB;       // 400
  mixer_kernel<<<blocks, WPB * 32, 0, stream>>>(
      qvals, states, a_h, gs,
      W1a, b1a, W1b, b1b, Wb1, bb1,
      Ww2a, bw2a, Ww2b, bw2b, Wb2a, bb2a, Wb2b, bb2b,
      W3a, b3a, W3b, b3b, Wb3, bb3,
      W4a, b4a, W4b, b4b, Wb4a, bb4a, Wb4b, bb4b,
      out_tot, out_sd, ws);
  w1avg_reduce<<<1, 32, 0, stream>>>(ws, out_w1, ntiles);
}